// H_Smooth_3848290697553
// MI455X (gfx1250) — compile-verified
//
#include <hip/hip_runtime.h>

typedef __attribute__((ext_vector_type(2))) float v2f;
typedef __attribute__((ext_vector_type(8))) float v8f;

#define NZ 4096
#define NX 8192
#define RADIUS 40
#define NTAPS (2 * RADIUS + 1)  // 81
#define KTOT 96                 // 81 taps + 15 intra-tile shift, padded to mult. of 4
#define NSTEP (KTOT / 4)        // 24 chained wmma_f32_16x16x4_f32 per output tile
#define ROWTILES 4              // 16x16 tiles per wave (amortizes B-tile build)

__device__ __forceinline__ int mirror_idx(int i) {
  // JAX 'symmetric' padding: reflect including the edge element.
  if (i < 0) i = -1 - i;
  if (i >= NX) i = 2 * NX - 1 - i;
  return i;
}

// Tiny init kernel: Gaussian taps into workspace (deterministic each call).
__global__ void gauss_weights_kernel(const int* __restrict__ h_smooth,
                                     float* __restrict__ w) {
  const int j = threadIdx.x;
  const float sigma = (float)h_smooth[0];
  const float inv2s2 = 0.5f / (sigma * sigma);
  float sum = 0.f;
  for (int t = 0; t < NTAPS; ++t) {
    float x = (float)(t - RADIUS);
    sum += __expf(-x * x * inv2s2);
  }
  if (j < NTAPS) {
    float x = (float)(j - RADIUS);
    w[j] = __expf(-x * x * inv2s2) / sum;
  }
}

// 256 threads = 8 waves. Each wave owns a 16-col strip and walks 4 row-tiles,
// reusing its Toeplitz B registers. Workgroup covers 64 rows x 128 cols.
__launch_bounds__(256)
__global__ void gauss_smooth_wmma(const float* __restrict__ in,
                                  const float* __restrict__ w,
                                  float* __restrict__ out) {
  __shared__ float ws[NTAPS];
  if (threadIdx.x < NTAPS) ws[threadIdx.x] = w[threadIdx.x];
  __syncthreads();

  const int lane = threadIdx.x & 31;
  const int wave = threadIdx.x >> 5;
  const int x0 = blockIdx.x * 128 + wave * 16;     // output column tile base
  const int rbase = blockIdx.y * (16 * ROWTILES);  // output row region base
  const int n = lane & 15;                         // N / M index this lane serves
  const int hi = lane >> 4;                        // lane-half (K split)

  // Banded Toeplitz B tile (K x 16): B[k][n] = w[k - n], identical for every
  // output tile. Operand layout per step: VGPR0 = rows k, k+2 ; VGPR1 = k+1, k+3.
  v2f B[NSTEP];
#pragma unroll
  for (int s = 0; s < NSTEP; ++s) {
    const int k = 4 * s + 2 * hi;
    const int d0 = k - n, d1 = k + 1 - n;
    B[s].x = (d0 >= 0 && d0 < NTAPS) ? ws[d0] : 0.f;
    B[s].y = (d1 >= 0 && d1 < NTAPS) ? ws[d1] : 0.f;
  }

  const int cbase = x0 - RADIUS;  // input column for k = 0
  const bool interior = (cbase >= 0) && (cbase + KTOT <= NX);

  for (int t = 0; t < ROWTILES; ++t) {
    const int r0 = rbase + t * 16;
    const float* inrow = in + (size_t)(r0 + n) * NX;
    v8f c = {};

    if (interior) {
      // Interior fast path: straight 8-byte vector loads (cbase is even -> 8B aligned).
#pragma unroll
      for (int s = 0; s < NSTEP; ++s) {
        const int col = cbase + 4 * s + 2 * hi;
        v2f a = *(const v2f*)(inrow + col);
        c = __builtin_amdgcn_wmma_f32_16x16x4_f32(false, a, false, B[s],
                                                  (short)0, c, false, false);
      }
    } else {
      // Boundary path (wave-uniform branch; EXEC stays all-ones for WMMA).
#pragma unroll
      for (int s = 0; s < NSTEP; ++s) {
        const int col = cbase + 4 * s + 2 * hi;
        v2f a;
        a.x = inrow[mirror_idx(col)];
        a.y = inrow[mirror_idx(col + 1)];
        c = __builtin_amdgcn_wmma_f32_16x16x4_f32(false, a, false, B[s],
                                                  (short)0, c, false, false);
      }
    }

    // C/D layout: VGPR r -> M = r (lanes 0-15) / r+8 (lanes 16-31), N = lane&15.
    // Each 16-float row is written by 16 consecutive lanes -> coalesced stores.
    float* orow = out + (size_t)(r0 + 8 * hi) * NX + x0 + n;
#pragma unroll
    for (int r = 0; r < 8; ++r) {
      orow[(size_t)r * NX] = c[r];
    }
  }
}

extern "C" void kernel_launch(void* const* d_in, const int* in_sizes, int n_in,
                              void* d_out, int out_size, void* d_ws, size_t ws_size,
                              hipStream_t stream) {
  (void)in_sizes; (void)n_in; (void)out_size; (void)ws_size;
  const float* feature = (const float*)d_in[0];
  const int* h_smooth = (const int*)d_in[1];
  float* w = (float*)d_ws;  // 81 floats of workspace

  gauss_weights_kernel<<<1, 128, 0, stream>>>(h_smooth, w);

  dim3 grid(NX / 128, NZ / (16 * ROWTILES));
  gauss_smooth_wmma<<<grid, 256, 0, stream>>>(feature, w, (float*)d_out);
}